// approx_dconv_34385508172476
// MI455X (gfx1250) — compile-verified
//
#include <hip/hip_runtime.h>
#include <stdint.h>

// Problem constants (match reference)
#define B_    8
#define C_    128
#define HW_   112
#define PLANE (HW_ * HW_)            // 12544 pixels per (b,c) plane
#define LUTN  256
#define LUT_BYTES (LUTN * LUTN * 4)  // 262144 B
#define X_BYTES   (PLANE * 4)        // 50176 B
#define QPR   (HW_ / 4)              // 28 quads per row
#define NQUAD (PLANE / 4)            // 3136 quads per plane
#define NTHR  512

typedef __attribute__((ext_vector_type(4))) unsigned int v4u;
typedef __attribute__((ext_vector_type(8))) int          v8i;
typedef __attribute__((ext_vector_type(4))) int          v4i;

// ---------------------------------------------------------------------------
// Tensor Data Mover: 1-D contiguous DMA Global -> LDS.
//   n8 = number of 8-byte elements (region 8B-aligned, multiple of 8B)
// D# group0: count=1 | lds_addr | global_addr[56:0] | type=2 ("image")
// D# group1: data_size=3 (8B), tensor_dim0=n8, tensor_dim1=1,
//            tile_dim0=n8 (1-D tile), tile_dim1/2=0, dim0_stride=n8
// ---------------------------------------------------------------------------
__device__ __forceinline__ void tdm_load_1d(unsigned lds_off, const void* gptr,
                                            unsigned n8)
{
    unsigned long long ga = (unsigned long long)(uintptr_t)gptr;
    v4u g0;
    g0.x = 1u;                                              // count=1, user desc
    g0.y = lds_off;                                         // lds_addr (bytes)
    g0.z = (unsigned)(ga & 0xFFFFFFFFull);                  // global_addr[31:0]
    g0.w = (unsigned)((ga >> 32) & 0x1FFFFFFull)            // global_addr[56:32]
         | (2u << 30);                                      // type=2
    v8i g1;
    g1[0] = (int)(3u << 16);                                // data_size=3 (8B)
    g1[1] = (int)((n8 & 0xFFFFu) << 16);                    // tensor_dim0[15:0]
    g1[2] = (int)((n8 >> 16) & 0xFFFFu) | (1 << 16);        // tensor_dim0[31:16], tensor_dim1=1
    g1[3] = (int)((n8 & 0xFFFFu) << 16);                    // tensor_dim1_hi=0, tile_dim0=n8
    g1[4] = 0;                                              // tile_dim1=0, tile_dim2=0
    g1[5] = (int)n8;                                        // tensor_dim0_stride[31:0]
    g1[6] = 0;
    g1[7] = 0;
    v4i z4 = {0, 0, 0, 0};
#if defined(__clang_major__) && (__clang_major__ >= 23)
    v8i z8 = {0, 0, 0, 0, 0, 0, 0, 0};
    __builtin_amdgcn_tensor_load_to_lds(g0, g1, z4, z4, z8, 0);
#else
    __builtin_amdgcn_tensor_load_to_lds(g0, g1, z4, z4, 0);
#endif
}

// ---------------------------------------------------------------------------
// One workgroup per (b,c) plane. LDS layout:
//   [0      .. 262143] : full 256x256 f32 LUT (TDM-staged)
//   [262144 .. 312319] : 112x112 f32 input plane (TDM-staged)
// 512 threads (16 wave32). Each thread iteration handles a quad of 4
// consecutive output pixels in one row: the 3x3 taps of the quad share a
// 3x6 input window (18 LDS x-reads feed 36 tap-pixels), quantization runs
// once per window value, and the 4 results go out as one b128 store.
// ---------------------------------------------------------------------------
__global__ void lut_dwconv_kernel(const float* __restrict__ x,
                                  const float* __restrict__ weight,
                                  const float* __restrict__ lut,
                                  float* __restrict__ out)
{
    extern __shared__ float smem[];
    float* lut_lds = smem;                  // 65536 floats
    float* x_lds   = smem + LUTN * LUTN;    // 12544 floats

    const int bc = blockIdx.x;              // = b*C + c
    const int c  = bc & (C_ - 1);
    const float* xpl = x + (size_t)bc * PLANE;

    // Wave 0 drives the TDM for both tiles, drains TENSORcnt, then the
    // workgroup barrier publishes the staged LDS to all 16 waves.
    if (threadIdx.x < 32) {
        tdm_load_1d((unsigned)(uintptr_t)lut_lds, lut, LUT_BYTES / 8);
        tdm_load_1d((unsigned)(uintptr_t)x_lds,   xpl, X_BYTES / 8);
        __builtin_amdgcn_s_wait_tensorcnt(0);
    }
    __syncthreads();

    // Per-channel quantized weights (uniform): LUT row offset + sign preds.
    const float* __restrict__ wc = weight + c * 9;
    int  rowoff[9];
    bool pos1[9], neg1[9];
#pragma unroll
    for (int k = 0; k < 9; ++k) {
        float t1 = rintf(wc[k] * 1000.0f);
        t1 = fminf(fmaxf(t1, -255.0f), 255.0f);
        pos1[k]   = (t1 > 0.0f);
        neg1[k]   = (t1 < 0.0f);
        rowoff[k] = ((int)fabsf(t1)) * LUTN;
    }

    float* opl = out + (size_t)bc * PLANE;

    for (int q = (int)threadIdx.x; q < NQUAD; q += NTHR) {
        const int oh  = q / QPR;
        const int ow0 = (q - oh * QPR) * 4;

        // Quantize the 3x6 input window once.
        int      idx[3][6];   // |t2| as LUT column index
        unsigned mp[3][6];    // t2>0 ? 0 : signbit
        unsigned mn[3][6];    // t2<0 ? 0 : signbit
#pragma unroll
        for (int r = 0; r < 3; ++r) {
            const int  ih = oh + r - 1;
            const bool rv = ((unsigned)ih < (unsigned)HW_);
#pragma unroll
            for (int j = 0; j < 6; ++j) {
                const int  iw = ow0 + j - 1;
                const bool v  = rv && ((unsigned)iw < (unsigned)HW_);
                // Padded taps still contribute -lut[|t1|,0] (sign rule), so
                // they flow through as t2 = 0.
                const float a = v ? x_lds[ih * HW_ + iw] : 0.0f;
                float t = rintf(a);
                t = fminf(fmaxf(t, -255.0f), 255.0f);
                idx[r][j] = (int)fabsf(t);
                mp[r][j]  = (t > 0.0f) ? 0u : 0x80000000u;
                mn[r][j]  = (t < 0.0f) ? 0u : 0x80000000u;
            }
        }

        float acc[4] = {0.0f, 0.0f, 0.0f, 0.0f};
#pragma unroll
        for (int ky = 0; ky < 3; ++ky) {
#pragma unroll
            for (int kx = 0; kx < 3; ++kx) {
                const int k = ky * 3 + kx;
#pragma unroll
                for (int j = 0; j < 4; ++j) {
                    const float    l  = lut_lds[rowoff[k] + idx[ky][j + kx]];
                    // sign = +1 iff (t1>0 && t2>0) || (t1<0 && t2<0); applied
                    // as one v_xor_b32 on the gathered bits.
                    const unsigned fm = pos1[k] ? mp[ky][j + kx]
                                     : (neg1[k] ? mn[ky][j + kx]
                                                : 0x80000000u);
                    acc[j] += __uint_as_float(__float_as_uint(l) ^ fm);
                }
            }
        }

        float4 o;
        o.x = acc[0] * 0.001f;
        o.y = acc[1] * 0.001f;
        o.z = acc[2] * 0.001f;
        o.w = acc[3] * 0.001f;
        *reinterpret_cast<float4*>(opl + oh * HW_ + ow0) = o;  // b128 store
    }
}

extern "C" void kernel_launch(void* const* d_in, const int* in_sizes, int n_in,
                              void* d_out, int out_size, void* d_ws, size_t ws_size,
                              hipStream_t stream)
{
    const float* x      = (const float*)d_in[0];   // (8,128,112,112)
    const float* weight = (const float*)d_in[1];   // (128,1,3,3)
    const float* lut    = (const float*)d_in[2];   // (256,256)
    float*       out    = (float*)d_out;           // (8,128,12544)

    const size_t smem_bytes = (size_t)LUT_BYTES + (size_t)X_BYTES; // 312320 < 320K
    (void)hipFuncSetAttribute((const void*)lut_dwconv_kernel,
                              hipFuncAttributeMaxDynamicSharedMemorySize,
                              (int)smem_bytes);

    dim3 grid(B_ * C_);   // 1024 workgroups, one (b,c) plane each
    dim3 block(NTHR);     // 16 wave32
    hipLaunchKernelGGL(lut_dwconv_kernel, grid, block, smem_bytes, stream,
                       x, weight, lut, out);
}